// RadiusGraph_81484119540284
// MI455X (gfx1250) — compile-verified
//
#include <hip/hip_runtime.h>

typedef __attribute__((ext_vector_type(2))) float v2f;
typedef __attribute__((ext_vector_type(8))) float v8f;

#define RG_RADIUS 0.08f

// One 16x16 tile: V_WMMA_F32_16X16X4_F32 with the -2 factor and the row norm
// folded into the A fragment (K=4 fully used):
//   A'_i = (-2x, -2y, -2z, |x_i|^2),  B'_j = (x_j, y_j, z_j, 1)
//   => acc[m][n] = |x_i|^2 - 2 x_i.x_j ; sq = acc + |x_j|^2 (one add/elem).
// `diag` is scalar-uniform and constant-folded at each call site.
__device__ __forceinline__ void rg_tile(int n, bool diag, v2f afrag,
                                        const float* __restrict__ qb,
                                        float* __restrict__ pm,
                                        float* __restrict__ pd,
                                        int hi, int t, int rowbase, int colbase)
{
    // column point (b96 load at immediate offset), norm, B fragment
    const float qx = qb[48 * t + 0];
    const float qy = qb[48 * t + 1];
    const float qz = qb[48 * t + 2];
    const float snb = qx * qx + qy * qy + qz * qz;
    v2f bfrag;
    bfrag.x = hi ? qz : qx;       // lanes 16-31: K=2 ; lanes 0-15: K=0
    bfrag.y = hi ? 1.0f : qy;     // lanes 16-31: K=3 ; lanes 0-15: K=1

    v8f acc = {};
    acc = __builtin_amdgcn_wmma_f32_16x16x4_f32(
        /*neg_a=*/false, afrag, /*neg_b=*/false, bfrag,
        /*c_mod=*/(short)0, acc, /*reuse_a=*/false, /*reuse_b=*/false);

    const int col = colbase + t * 16;   // global column this lane serves
#pragma unroll
    for (int r = 0; r < 8; ++r) {
        const float sq   = fmaxf(acc[r] + snb, 0.0f);
        const float dist = __builtin_amdgcn_sqrtf(sq);   // v_sqrt_f32, ~1 ULP
        bool edge = dist <= RG_RADIUS;
        if (diag) edge = edge && ((rowbase + r) != col); // diagonal waves only
        const size_t off = (size_t)r * (size_t)n + (size_t)(t * 16);
        __builtin_nontemporal_store(edge ? 1.0f : 0.0f, pm + off);
        __builtin_nontemporal_store(edge ? dist : 0.0f, pd + off);
    }
}

// NS != 0: compile-time matrix dimension -> all store/load offsets become
// immediate fields off a single base address per plane. NS == 0: runtime n.
template <int NS, int TJ>
__global__ __launch_bounds__(256) void radius_graph_wmma_kernel(
    const float* __restrict__ x,
    float* __restrict__ out_mask,
    float* __restrict__ out_dist,
    int nrt)
{
    const int n    = NS ? NS : nrt;
    const int lane = threadIdx.x & 31;
    const int lr   = lane & 15;   // row/col within tile this lane serves
    const int hi   = lane >> 4;   // wave half (K-split for A/B, M-split for D)

    // Force the wave id into an SGPR so tile indexing and the diagonal
    // dispatch are scalar: branches below are s_cbranch (EXEC never touched,
    // satisfying the WMMA all-ones EXEC requirement by construction).
    const int wave = __builtin_amdgcn_readfirstlane((int)(threadIdx.x >> 5));

    const int tiles   = n >> 4;
    const int groupsJ = tiles / TJ;
    const long long tid = (long long)blockIdx.x * 8 + wave;
    const int ti = (int)(tid / groupsJ);
    const int tg = (int)(tid % groupsJ);
    if (ti >= tiles) return;      // scalar-uniform guard

    const int i0 = ti * 16;
    const int j0 = tg * (16 * TJ);

    // ---- A fragment with folded -2 and row norm
    const int irow = i0 + lr;
    const float px = x[3 * irow + 0];
    const float py = x[3 * irow + 1];
    const float pz = x[3 * irow + 2];
    const float sni = px * px + py * py + pz * pz;
    v2f afrag;
    afrag.x = hi ? (-2.0f * pz) : (-2.0f * px);
    afrag.y = hi ? sni          : (-2.0f * py);

    // ---- one base address per plane; every store uses immediate offsets
    const int rowbase  = i0 + hi * 8;               // D layout: VGPR r -> M = r + 8*hi
    const int colbase  = j0 + lr;
    const size_t obase = (size_t)rowbase * (size_t)n + (size_t)colbase;
    float* const pm = out_mask + obase;
    float* const pd = out_dist + obase;
    const float* const qb = x + 3 * (size_t)colbase;

    // ---- diagonal intersects only 1/groupsJ of wave-units: scalar branch
    const bool hasDiag = (ti / TJ) == tg;
    if (hasDiag) {
#pragma unroll
        for (int t = 0; t < TJ; ++t)
            rg_tile(n, true, afrag, qb, pm, pd, hi, t, rowbase, colbase);
    } else {
#pragma unroll
        for (int t = 0; t < TJ; ++t)
            rg_tile(n, false, afrag, qb, pm, pd, hi, t, rowbase, colbase);
    }
}

extern "C" void kernel_launch(void* const* d_in, const int* in_sizes, int n_in,
                              void* d_out, int out_size, void* d_ws, size_t ws_size,
                              hipStream_t stream) {
    (void)n_in; (void)d_ws; (void)ws_size; (void)out_size;

    const float* x = (const float*)d_in[0];
    const int n = in_sizes[0] / 3;                 // 16384 points, D=3

    float* out_mask = (float*)d_out;
    float* out_dist = out_mask + (size_t)n * (size_t)n;

    constexpr int TJ = 4;                          // column tiles per wave
    const int tiles = n >> 4;                      // 1024
    const long long waveUnits = (long long)tiles * (tiles / TJ);  // 262144
    const long long blocks = (waveUnits + 7) / 8;  // 8 waves (256 thr) / block

    if (n == 16384) {
        radius_graph_wmma_kernel<16384, TJ>
            <<<dim3((unsigned)blocks), dim3(256), 0, stream>>>(x, out_mask, out_dist, n);
    } else {
        radius_graph_wmma_kernel<0, TJ>
            <<<dim3((unsigned)blocks), dim3(256), 0, stream>>>(x, out_mask, out_dist, n);
    }
}